// GraphSage_23940147708459
// MI455X (gfx1250) — compile-verified
//
#include <hip/hip_runtime.h>
#include <hip/hip_bf16.h>
#include <math.h>

#define D      128
#define WPITCH 288   // floats per k-pair row of a weight matrix in LDS
                     // (2*128 cols interleaved + 32 pad: lanes16-31 offset
                     //  288%64==32 banks -> conflict-free ds_load_b64 frags)
#define WSZ    (64 * WPITCH)  // 18432 floats per 128x128 weight matrix
#define ALD    129   // LDS row stride for activations (conflict-free A frags)
#define ASZ    (64 * ALD)

typedef __attribute__((ext_vector_type(2))) float v2f;
typedef __attribute__((ext_vector_type(8))) float v8f;

__device__ __forceinline__ v8f wmma_f32(v2f a, v2f b, v8f c) {
  // V_WMMA_F32_16X16X4_F32 : D = A(16x4 f32) x B(4x16 f32) + C(16x16 f32)
  return __builtin_amdgcn_wmma_f32_16x16x4_f32(
      /*neg_a=*/false, a, /*neg_b=*/false, b,
      /*c_mod=*/(short)0, c, /*reuse_a=*/false, /*reuse_b=*/false);
}

// ---------------------------------------------------------------------------
// dst[sidx[e]] += src[gidx[e]]  over 128-float rows. One wave32 per edge,
// float4 gather per lane + 4 global_atomic_add_f32.
// ---------------------------------------------------------------------------
__global__ __launch_bounds__(256) void scatter_add_kernel(
    const float* __restrict__ src, const int* __restrict__ gidx,
    const int* __restrict__ sidx, float* __restrict__ dst, int n) {
  int e = (int)((blockIdx.x * (unsigned)blockDim.x + threadIdx.x) >> 5);
  if (e >= n) return;
  int lane = threadIdx.x & 31;
  size_t g = (size_t)gidx[e] * D + (size_t)lane * 4;
  size_t s = (size_t)sidx[e] * D + (size_t)lane * 4;
  float4 v = *(const float4*)(src + g);
  float* d = dst + s;
  atomicAdd(d + 0, v.x);
  atomicAdd(d + 1, v.y);
  atomicAdd(d + 2, v.z);
  atomicAdd(d + 3, v.w);
}

// ---------------------------------------------------------------------------
// Fused SAGE update:  Y = X0*W0 (+ X1*W1 + X2*W2) + (b0(+b1+b2))
// then optional L2 row-normalize (eps clamp) and leaky-relu.
// Block = 128 threads (4 waves), 64 rows; wave w owns rows [row0+16w,+16)
// x all 128 output columns (8 WMMA column tiles, v8f accumulators).
// Weights stored in LDS in k-paired layout: addr(k,n)=(k>>1)*WPITCH+2n+(k&1)
// so each B fragment {W[k][n],W[k+1][n]} is ONE aligned ds_load_b64 landing
// in a contiguous even VGPR pair (no re-pairing movs). All LDS accesses are
// smem[int_index] to stay in AS(3) (ds_*, not flat_*).
// ---------------------------------------------------------------------------
template <int NMAT>
__global__ __launch_bounds__(128) void sage_gemm(
    const float* __restrict__ X0, const float* __restrict__ X1,
    const float* __restrict__ X2,
    const float* __restrict__ W0, const float* __restrict__ W1,
    const float* __restrict__ W2,
    const float* __restrict__ b0, const float* __restrict__ b1,
    const float* __restrict__ b2,
    float* __restrict__ Y, int M, int doNorm, int doAct) {
  extern __shared__ float smem[];
  const int sAbase = NMAT * WSZ;         // weights at 0, activations after
  const int tid  = threadIdx.x;
  const int row0 = blockIdx.x * 64;

  const float* Xg[3] = {X0, X1, X2};
  const float* Wg[3] = {W0, W1, W2};
  const float* bg[3] = {b0, b1, b2};

  // ---- stage weights: row-major 128x128 -> k-paired LDS layout ----
  for (int t = 0; t < NMAT; ++t) {
    const float* w = Wg[t];
    const int base = t * WSZ;
    for (int idx = tid; idx < D * (D / 4); idx += 128) {
      int k = idx >> 5;
      int c = (idx & 31) << 2;
      float4 v = *(const float4*)(w + k * D + c);
      const int o = base + (k >> 1) * WPITCH + (k & 1) + 2 * c;
      smem[o + 0] = v.x; smem[o + 2] = v.y;
      smem[o + 4] = v.z; smem[o + 6] = v.w;
    }
  }
  // ---- stage 64 activation rows per input (clamped for tail) ----
  for (int t = 0; t < NMAT; ++t) {
    const float* x = Xg[t];
    const int base = sAbase + t * ASZ;
    for (int idx = tid; idx < 64 * (D / 4); idx += 128) {
      int r = idx >> 5;
      int c = (idx & 31) << 2;
      int gr = row0 + r;
      if (gr >= M) gr = M - 1;
      float4 v = *(const float4*)(x + (size_t)gr * D + c);
      const int o = base + r * ALD + c;
      smem[o + 0] = v.x; smem[o + 1] = v.y;
      smem[o + 2] = v.z; smem[o + 3] = v.w;
    }
  }
  __syncthreads();

  const int wid  = tid >> 5;     // wave id == row tile
  const int lane = tid & 31;
  const int half = lane >> 4;    // 0: lanes 0-15, 1: lanes 16-31
  const int lq   = lane & 15;

  if (row0 + wid * 16 >= M) return;   // wave-uniform: EXEC stays all-1s

  v8f acc[8];
  for (int nt = 0; nt < 8; ++nt)
    for (int j = 0; j < 8; ++j) acc[nt][j] = 0.0f;

  // Loop-carried integer indices; const parts fold into ds offsets.
  int aidx[NMAT];   // A frag: row (wid*16+lq), k = 2*half (+4 per kt)
  int widx[NMAT];   // B frag base: kpair = half (+2 per kt), col lq
#pragma unroll
  for (int t = 0; t < NMAT; ++t) {
    aidx[t] = sAbase + t * ASZ + (wid * 16 + lq) * ALD + 2 * half;
    widx[t] = t * WSZ + half * WPITCH + 2 * lq;
  }

#pragma unroll 2
  for (int kt = 0; kt < 32; ++kt) {            // K = 128 in steps of 4
    v2f a[NMAT];
    v2f bf[NMAT][8];
#pragma unroll
    for (int t = 0; t < NMAT; ++t) {           // ds_load_2addr offset0:0/1
      a[t].x = smem[aidx[t]];
      a[t].y = smem[aidx[t] + 1];
    }
#pragma unroll
    for (int t = 0; t < NMAT; ++t)
#pragma unroll
      for (int nt = 0; nt < 8; ++nt) {         // one ds_load_b64 per frag
        bf[t][nt].x = smem[widx[t] + nt * 32];
        bf[t][nt].y = smem[widx[t] + nt * 32 + 1];
      }
#pragma unroll
    for (int t = 0; t < NMAT; ++t)
#pragma unroll
      for (int nt = 0; nt < 8; ++nt)           // 8*NMAT back-to-back WMMAs
        acc[nt] = wmma_f32(a[t], bf[t][nt], acc[nt]);
#pragma unroll
    for (int t = 0; t < NMAT; ++t) {
      aidx[t] += 4;                            // next k-group (k += 4)
      widx[t] += 2 * WPITCH;                   // two k-pair rows
    }
  }

  // ---- bias ----
#pragma unroll
  for (int nt = 0; nt < 8; ++nt) {
    const int n = nt * 16 + lq;
    float bb = 0.0f;
    for (int t = 0; t < NMAT; ++t) bb += bg[t][n];
    for (int j = 0; j < 8; ++j) acc[nt][j] += bb;
  }

  // ---- L2 row-normalize: row m=r+8*half lives across the 16 lanes of a half ----
  if (doNorm) {
#pragma unroll
    for (int r = 0; r < 8; ++r) {
      float ss = 0.0f;
      for (int nt = 0; nt < 8; ++nt) ss += acc[nt][r] * acc[nt][r];
      for (int m = 1; m < 16; m <<= 1) ss += __shfl_xor(ss, m, 32);
      float sc = 1.0f / fmaxf(sqrtf(ss), 1e-12f);
      for (int nt = 0; nt < 8; ++nt) acc[nt][r] *= sc;
    }
  }
  if (doAct) {
    for (int nt = 0; nt < 8; ++nt)
      for (int j = 0; j < 8; ++j) {
        float v = acc[nt][j];
        acc[nt][j] = (v >= 0.0f) ? v : 0.01f * v;
      }
  }

  // ---- store C/D layout: VGPR r, lane L -> row r+8*(L>>4), col (L&15)+16*nt ----
#pragma unroll
  for (int r = 0; r < 8; ++r) {
    int gm = row0 + wid * 16 + r + 8 * half;
    if (gm < M) {
#pragma unroll
      for (int nt = 0; nt < 8; ++nt)
        Y[(size_t)gm * D + nt * 16 + lq] = acc[nt][r];
    }
  }
}

// ---------------------------------------------------------------------------
// Column max pooling (2 stage)
// ---------------------------------------------------------------------------
__global__ __launch_bounds__(128) void colmax_kernel(
    const float* __restrict__ X, int M, float* __restrict__ part) {
  int c = threadIdx.x;
  float m = -3.402823466e38f;
  for (long long r = blockIdx.x; r < (long long)M; r += gridDim.x)
    m = fmaxf(m, X[r * (long long)D + c]);
  part[(size_t)blockIdx.x * D + c] = m;
}

__global__ __launch_bounds__(128) void pool_finish_kernel(
    const float* __restrict__ pn, const float* __restrict__ pe, int P,
    float* __restrict__ out) {
  int c = threadIdx.x;
  float mn = -3.402823466e38f, me = -3.402823466e38f;
  for (int i = 0; i < P; ++i) {
    mn = fmaxf(mn, pn[(size_t)i * D + c]);
    me = fmaxf(me, pe[(size_t)i * D + c]);
  }
  out[c] = mn + me;
}

// ---------------------------------------------------------------------------
extern "C" void kernel_launch(void* const* d_in, const int* in_sizes, int n_in,
                              void* d_out, int out_size, void* d_ws,
                              size_t ws_size, hipStream_t stream) {
  const float* node_feature = (const float*)d_in[0];
  const float* edge_feature = (const float*)d_in[1];
  const float* node_Wc = (const float*)d_in[2];
  const float* node_bc = (const float*)d_in[3];
  const float* node_Wn = (const float*)d_in[4];
  const float* node_bn = (const float*)d_in[5];
  const float* node_We = (const float*)d_in[6];
  const float* node_be = (const float*)d_in[7];
  const float* edge_Wc = (const float*)d_in[8];
  const float* edge_bc = (const float*)d_in[9];
  const float* edge_Wn = (const float*)d_in[10];
  const float* edge_bn = (const float*)d_in[11];
  const float* edge_We = (const float*)d_in[12];
  const float* edge_be = (const float*)d_in[13];
  const float* W_node_out = (const float*)d_in[14];
  const float* b_node_out = (const float*)d_in[15];
  const float* W_edge_out = (const float*)d_in[16];
  const float* b_edge_out = (const float*)d_in[17];
  const int* edge_index        = (const int*)d_in[18];  // [2,E]
  const int* line_edge_index   = (const int*)d_in[19];  // [2,L]
  const int* node_edge_index   = (const int*)d_in[20];  // [E]
  const int* node_edge_scatter = (const int*)d_in[21];  // [E]
  const int* edge_node_index   = (const int*)d_in[22];  // [L]
  const int* edge_node_scatter = (const int*)d_in[23];  // [L]

  const int N = in_sizes[0] / D;
  const int E = in_sizes[1] / D;
  const int L = in_sizes[19] / 2;

  // workspace partition
  float* ws = (float*)d_ws;
  size_t off = 0;
  float* nfA  = ws + off; off += (size_t)N * D;
  float* nfB  = ws + off; off += (size_t)N * D;
  float* efA  = ws + off; off += (size_t)E * D;
  float* efB  = ws + off; off += (size_t)E * D;
  float* agN  = ws + off; off += (size_t)N * D;
  float* ag1N = ws + off; off += (size_t)N * D;
  float* agE  = ws + off; off += (size_t)E * D;
  float* ag1E = ws + off; off += (size_t)E * D;
  const int P = 512;
  float* pn = ws + off; off += (size_t)P * D;
  float* pe = ws + off; off += (size_t)P * D;

  const size_t shW = (size_t)WSZ * sizeof(float);     // 73728 B
  const size_t shA = (size_t)ASZ * sizeof(float);     // 33024 B
  const size_t sh3 = 3 * (shW + shA);   // 320256 B  (< 320 KB WGP LDS)
  const size_t sh1 = shW + shA;

  const float* nf = node_feature;
  const float* ef = edge_feature;
  float* nfNext = nfA;
  float* efNext = efA;

  const unsigned gScE = (unsigned)(((size_t)E * 32 + 255) / 256);
  const unsigned gScL = (unsigned)(((size_t)L * 32 + 255) / 256);
  const unsigned gN   = (unsigned)(((N + 15) / 16 + 3) / 4);
  const unsigned gE   = (unsigned)(((E + 15) / 16 + 3) / 4);

  for (int i = 0; i < 3; ++i) {
    const int act = (i < 2) ? 1 : 0;
    hipMemsetAsync(agN,  0, (size_t)N * D * sizeof(float), stream);
    hipMemsetAsync(ag1N, 0, (size_t)N * D * sizeof(float), stream);
    hipMemsetAsync(agE,  0, (size_t)E * D * sizeof(float), stream);
    hipMemsetAsync(ag1E, 0, (size_t)E * D * sizeof(float), stream);

    // node aggregations (read pre-update nf/ef)
    scatter_add_kernel<<<gScE, 256, 0, stream>>>(nf, edge_index, edge_index + E, agN, E);
    scatter_add_kernel<<<gScE, 256, 0, stream>>>(ef, node_edge_index, node_edge_scatter, ag1N, E);
    // edge aggregations (read pre-update ef/nf)
    scatter_add_kernel<<<gScL, 256, 0, stream>>>(ef, line_edge_index, line_edge_index + L, agE, L);
    scatter_add_kernel<<<gScL, 256, 0, stream>>>(nf, edge_node_index, edge_node_scatter, ag1E, L);

    sage_gemm<3><<<gN, 128, sh3, stream>>>(
        nf, agN, ag1N,
        node_Wc + (size_t)i * D * D, node_Wn + (size_t)i * D * D, node_We + (size_t)i * D * D,
        node_bc + (size_t)i * D, node_bn + (size_t)i * D, node_be + (size_t)i * D,
        nfNext, N, /*norm=*/1, act);
    sage_gemm<3><<<gE, 128, sh3, stream>>>(
        ef, agE, ag1E,
        edge_Wc + (size_t)i * D * D, edge_Wn + (size_t)i * D * D, edge_We + (size_t)i * D * D,
        edge_bc + (size_t)i * D, edge_bn + (size_t)i * D, edge_be + (size_t)i * D,
        efNext, E, /*norm=*/1, act);

    nf = nfNext;  ef = efNext;
    nfNext = (nfNext == nfA) ? nfB : nfA;
    efNext = (efNext == efA) ? efB : efA;
  }

  // final linears straight into d_out (pooled[128] | tn[N*128] | te[E*128])
  float* out = (float*)d_out;
  float* tn  = out + D;
  float* te  = out + D + (size_t)N * D;
  sage_gemm<1><<<gN, 128, sh1, stream>>>(
      nf, nullptr, nullptr, W_node_out, nullptr, nullptr,
      b_node_out, nullptr, nullptr, tn, N, 0, 0);
  sage_gemm<1><<<gE, 128, sh1, stream>>>(
      ef, nullptr, nullptr, W_edge_out, nullptr, nullptr,
      b_edge_out, nullptr, nullptr, te, E, 0, 0);

  colmax_kernel<<<P, D, 0, stream>>>(tn, N, pn);
  colmax_kernel<<<P, D, 0, stream>>>(te, E, pe);
  pool_finish_kernel<<<1, D, 0, stream>>>(pn, pe, P, out);
}